// EmbeddingModel_29394756174316
// MI455X (gfx1250) — compile-verified
//
#include <hip/hip_runtime.h>
#include <math.h>

// word2vec-style embedding loss for MI455X (gfx1250, wave32).
// Memory-bound (~516 MB streamed, ~252 useful MFLOP). WMMA f32 16x16x4 is used
// as a "diagonal" engine: 16 independent 128-dim f32 dot products per wave tile,
// with the K-reduction done by the matrix unit.

typedef float v2f __attribute__((ext_vector_type(2)));
typedef float v8f __attribute__((ext_vector_type(8)));

#define LE_LAMBDA 1e-8f

__device__ __forceinline__ float log_sigmoid_stable(float x) {
    // log(sigmoid(x)) = min(x,0) - log1p(exp(-|x|))
    return fminf(x, 0.0f) - log1pf(expf(-fabsf(x)));
}

__global__ void init_accum_kernel(float* acc) {
    if (threadIdx.x < 2) acc[threadIdx.x] = 0.0f;
}

// accum[0] += sum over all (b,j) of log_sigmoid(sign * dot(out_row, in_row))
__global__ void graph_loss_kernel(const int* __restrict__ input_labels,
                                  const int* __restrict__ pos_labels,
                                  const int* __restrict__ neg_labels,
                                  const float* __restrict__ in_w,
                                  const float* __restrict__ out_w,
                                  float* __restrict__ accum,
                                  int Bn, int Cc, int Nn) {
    const int JT = Cc + Nn;                      // 60
    const int total_pairs = Bn * JT;             // 983040
    const int num_tiles = (total_pairs + 15) >> 4;

    const int lane  = threadIdx.x & 31;
    const int m     = lane & 15;                 // which of the 16 pairs in the tile
    const int half  = lane >> 4;                 // K half: 0 -> {k,k+1}, 1 -> {k+2,k+3}
    const int wave  = (int)((blockIdx.x * blockDim.x + threadIdx.x) >> 5);
    const int nwav  = (int)((gridDim.x * blockDim.x) >> 5);

    float local = 0.0f;

    for (int t = wave; t < num_tiles; t += nwav) {   // wave-uniform loop (EXEC stays full)
        int p = t * 16 + m;
        bool active = p < total_pairs;
        int pc = active ? p : 0;
        int b = pc / JT;
        int j = pc - b * JT;

        int   label;
        float sign;
        if (j < Cc) { label = pos_labels[b * Cc + j];        sign =  1.0f; }
        else        { label = neg_labels[b * Nn + (j - Cc)]; sign = -1.0f; }

        const float* rowO = out_w + (long)label * 128;
        const float* rowI = in_w  + (long)input_labels[b] * 128;
        __builtin_prefetch(rowO, 0, 0);
        __builtin_prefetch(rowI, 0, 0);

        v8f acc = {0.f, 0.f, 0.f, 0.f, 0.f, 0.f, 0.f, 0.f};
        #pragma unroll
        for (int k = 0; k < 128; k += 4) {
            // A: 16x4 f32 — lane m holds K{k,k+1}, lane m+16 holds K{k+2,k+3}
            v2f av = *(const v2f*)(rowO + k + 2 * half);
            // B: 4x16 f32 — column n = lane&15 comes from pair n's input row
            v2f bv = *(const v2f*)(rowI + k + 2 * half);
            acc = __builtin_amdgcn_wmma_f32_16x16x4_f32(
                false, av, false, bv, (short)0, acc, false, false);
        }

        // Diagonal D[m][m]: lanes 0..7 -> acc[lane]; lanes 24..31 -> acc[lane-24].
        float dval = 0.0f;
        #pragma unroll
        for (int r = 0; r < 8; ++r) {
            float c = acc[r];
            dval = (lane == r)      ? c : dval;
            dval = (lane == 24 + r) ? c : dval;
        }
        bool is_diag = (lane < 8) || (lane >= 24);
        if (active && is_diag) {
            local += log_sigmoid_stable(sign * dval);
        }
    }

    // wave32 reduction, one atomic per wave
    #pragma unroll
    for (int off = 16; off > 0; off >>= 1)
        local += __shfl_down(local, off, 32);
    if (lane == 0) atomicAdd(&accum[0], local);
}

// accum[1] += sum_p ||in_w[pair0] - in_w[pair1]||^2
__global__ void hier_loss_kernel(const int* __restrict__ pairs,
                                 const float* __restrict__ in_w,
                                 float* __restrict__ accum,
                                 int Pp) {
    const long total = (long)Pp * 128;
    float s = 0.0f;
    for (long i = (long)blockIdx.x * blockDim.x + threadIdx.x; i < total;
         i += (long)gridDim.x * blockDim.x) {
        long pi = i >> 7;
        int  dp = (int)(i & 127);
        int a = pairs[pi * 2 + 0];
        int c = pairs[pi * 2 + 1];
        float diff = in_w[(long)a * 128 + dp] - in_w[(long)c * 128 + dp];
        s += diff * diff;
    }
    #pragma unroll
    for (int off = 16; off > 0; off >>= 1)
        s += __shfl_down(s, off, 32);
    if ((threadIdx.x & 31) == 0) atomicAdd(&accum[1], s);
}

__global__ void finalize_kernel(const float* __restrict__ accum,
                                float* __restrict__ out, int Bn) {
    if (threadIdx.x == 0 && blockIdx.x == 0) {
        float lh = 0.5f * LE_LAMBDA * accum[1];
        float loss_graph = -(accum[0] / (float)Bn);
        out[0] = loss_graph + lh;
        out[1] = lh;
    }
}

extern "C" void kernel_launch(void* const* d_in, const int* in_sizes, int n_in,
                              void* d_out, int out_size, void* d_ws, size_t ws_size,
                              hipStream_t stream) {
    const int*   input_labels = (const int*)d_in[0];
    const int*   pos_labels   = (const int*)d_in[1];
    const int*   neg_labels   = (const int*)d_in[2];
    const int*   hier_pairs   = (const int*)d_in[3];
    const float* in_w         = (const float*)d_in[4];
    const float* out_w        = (const float*)d_in[5];
    float*       out          = (float*)d_out;
    float*       accum        = (float*)d_ws;   // [0]=graph logsum, [1]=hier sq-sum

    const int Bn = in_sizes[0];
    const int Cc = in_sizes[1] / Bn;   // 10
    const int Nn = in_sizes[2] / Bn;   // 50
    const int Pp = in_sizes[3] / 2;    // 4096

    init_accum_kernel<<<1, 32, 0, stream>>>(accum);
    graph_loss_kernel<<<1024, 256, 0, stream>>>(input_labels, pos_labels, neg_labels,
                                                in_w, out_w, accum, Bn, Cc, Nn);
    hier_loss_kernel<<<512, 256, 0, stream>>>(hier_pairs, in_w, accum, Pp);
    finalize_kernel<<<1, 1, 0, stream>>>(accum, out, Bn);
}